// GCN_40252433498207
// MI455X (gfx1250) — compile-verified
//
#include <hip/hip_runtime.h>
#include <cstdint>

typedef __attribute__((ext_vector_type(2))) float v2f;
typedef __attribute__((ext_vector_type(8))) float v8f;

#define HH 32  // hidden width

// ---------------- degree / norm pipeline ----------------
__global__ void deg_init_k(float* deg, int n) {
  int i = blockIdx.x * blockDim.x + threadIdx.x;
  if (i < n) deg[i] = 1.0f;   // self loop
}

__global__ void deg_count_k(const long long* __restrict__ ei, float* deg, long long E) {
  long long e = (long long)blockIdx.x * blockDim.x + threadIdx.x;
  if (e < E) {
    long long d = ei[E + e];  // dst row of edge_index
    unsafeAtomicAdd(&deg[d], 1.0f);
  }
}

__global__ void deg_fin_k(float* deg, int n) {
  int i = blockIdx.x * blockDim.x + threadIdx.x;
  if (i < n) deg[i] = rsqrtf(deg[i]);  // deg >= 1 always
}

// ---------------- GCN layer GEMM: H = X[K] @ W[K,32], fused self-loop+bias init ----
// W staged in LDS *pair-swizzled* so each lane's B fragment (k,k+1 for one column)
// is one aligned ds_load_b64 straight into an even VGPR pair (no mov shuffles).
// One wave per 16-row M tile; two 16x16 WMMA accumulators; K loop step 4.
// Writes Hraw = X@W (for edge gather) and Hagg = bias + dinv^2 * Hraw (agg init).
template <int KDIM>
__global__ void gcn_gemm_wmma(const float* X,                      // [n, KDIM] (may alias Hagg)
                              const float* __restrict__ W,         // [KDIM, 32]
                              const float* __restrict__ bias,      // [32]
                              const float* __restrict__ dinv,      // [n]
                              float* __restrict__ Hraw,            // [n, 32]
                              float* Hagg,                         // [n, 32]
                              int n) {
  __shared__ __align__(8) float Wl[KDIM * HH];   // pair-swizzled weights, <= 16 KB
  __shared__ float Bl[HH];
  for (int i = threadIdx.x; i < KDIM * HH; i += blockDim.x) {
    const int k = i >> 5;        // row (k) in W
    const int c = i & 31;        // col (n) in W
    // pair-swizzled: (k/2) pair-row, then [col][k&1]
    Wl[((k >> 1) << 6) + (c << 1) + (k & 1)] = W[i];
  }
  if (threadIdx.x < HH) Bl[threadIdx.x] = bias[threadIdx.x];
  __syncthreads();

  const int lane  = threadIdx.x & 31;
  const int wave  = (blockIdx.x * blockDim.x + threadIdx.x) >> 5;
  const int m0    = wave * 16;
  if (m0 >= n) return;                       // wave-uniform: EXEC stays all-ones
  const int mrow  = m0 + (lane & 15);        // A: lane 0-15 / 16-31 both hold M=0..15
  const int khalf = (lane >> 4) << 1;        // lanes 16-31 hold K+2,K+3
  const int ncol  = lane & 15;               // B/C/D column within 16-wide tile

  const v2f* Wp = (const v2f*)Wl;            // pair view: Wp[pairRow*32 + col]

  v8f c0 = {};
  v8f c1 = {};
#pragma unroll
  for (int k0 = 0; k0 < KDIM; k0 += 4) {
    v2f a, b0, b1;
    const float* xr = X + (size_t)mrow * KDIM + k0 + khalf;
    a.x = xr[0];
    a.y = xr[1];
    const int pb = ((k0 + khalf) >> 1) << 5;  // pair-row base
    b0 = Wp[pb + ncol];                       // single ds_load_b64
    b1 = Wp[pb + ncol + 16];
    c0 = __builtin_amdgcn_wmma_f32_16x16x4_f32(false, a, false, b0, (short)0, c0, false, false);
    c1 = __builtin_amdgcn_wmma_f32_16x16x4_f32(false, a, false, b1, (short)0, c1, false, false);
  }

  // C/D layout: VGPR r -> M = r + (lane>=16 ? 8 : 0), N = lane&15
  const int mbase = m0 + ((lane >> 4) << 3);
  const float bn0 = Bl[ncol];
  const float bn1 = Bl[ncol + 16];
#pragma unroll
  for (int r = 0; r < 8; ++r) {
    const int m = mbase + r;
    const float di = dinv[m];
    const float d2 = di * di;
    const float v0 = c0[r];
    const float v1 = c1[r];
    const size_t o = (size_t)m * HH + ncol;
    Hraw[o]      = v0;
    Hraw[o + 16] = v1;
    Hagg[o]      = bn0 + d2 * v0;  // self-loop term + bias
    Hagg[o + 16] = bn1 + d2 * v1;
  }
}

// ---------------- edge scatter: one edge per wave, one feature per lane -------------
// Edge id forced scalar -> edge record & dinv via SMEM; feature access via saddr form.
__global__ void edge_scatter_k(const long long* __restrict__ ei,
                               const float* __restrict__ dinv,
                               const float* __restrict__ hin,   // raw h
                               float* __restrict__ hout,        // agg accumulator
                               long long E) {
  unsigned tid = blockIdx.x * blockDim.x + threadIdx.x;
  unsigned e = (unsigned)__builtin_amdgcn_readfirstlane(tid >> 5);  // wave-uniform edge id
  if (e >= (unsigned long long)E) return;
  const int f = threadIdx.x & 31;
  const long long s = ei[e];
  const long long d = ei[E + e];
  const float nrm = dinv[s] * dinv[d];
  const float v = hin[s * HH + f] * nrm;
  unsafeAtomicAdd(&hout[d * HH + f], v);
}

// ---------------- elementwise sigmoid (float4 vectorized) ----------------
__global__ void sigmoid_k(float4* __restrict__ h, long long n4) {
  long long i = (long long)blockIdx.x * blockDim.x + threadIdx.x;
  if (i < n4) {
    float4 v = h[i];
    v.x = 1.0f / (1.0f + __expf(-v.x));
    v.y = 1.0f / (1.0f + __expf(-v.y));
    v.z = 1.0f / (1.0f + __expf(-v.z));
    v.w = 1.0f / (1.0f + __expf(-v.w));
    h[i] = v;
  }
}

// ---------------- output head: out = sigmoid(X[n,32] @ Wlin[32,128] + blin) --------
__global__ void out_gemm_wmma(const float* __restrict__ X,     // [n, 32]
                              const float* __restrict__ W,     // [32, 128]
                              const float* __restrict__ bias,  // [128]
                              float* __restrict__ out,         // [n, 128]
                              int n) {
  __shared__ __align__(8) float Wl[HH * 128];   // pair-swizzled, 16 KB
  __shared__ float Bl[128];
  for (int i = threadIdx.x; i < HH * 128; i += blockDim.x) {
    const int k = i >> 7;        // row (k) in W
    const int c = i & 127;       // col (n) in W
    Wl[((k >> 1) << 8) + (c << 1) + (k & 1)] = W[i];
  }
  if (threadIdx.x < 128) Bl[threadIdx.x] = bias[threadIdx.x];
  __syncthreads();

  const int lane  = threadIdx.x & 31;
  const int wave  = (blockIdx.x * blockDim.x + threadIdx.x) >> 5;
  const int m0    = wave * 16;
  if (m0 >= n) return;
  const int mrow  = m0 + (lane & 15);
  const int khalf = (lane >> 4) << 1;
  const int ncol  = lane & 15;

  const v2f* Wp = (const v2f*)Wl;            // pair view: Wp[pairRow*128 + col]

  v8f acc[8] = {};
#pragma unroll
  for (int k0 = 0; k0 < HH; k0 += 4) {
    v2f a;
    const float* xr = X + (size_t)mrow * HH + k0 + khalf;
    a.x = xr[0];
    a.y = xr[1];
    const int pb = ((k0 + khalf) >> 1) << 7;  // pair-row base
#pragma unroll
    for (int t = 0; t < 8; ++t) {
      v2f b = Wp[pb + t * 16 + ncol];         // single ds_load_b64
      acc[t] = __builtin_amdgcn_wmma_f32_16x16x4_f32(false, a, false, b, (short)0, acc[t], false, false);
    }
  }

  const int mbase = m0 + ((lane >> 4) << 3);
#pragma unroll
  for (int t = 0; t < 8; ++t) {
    const float bn = Bl[t * 16 + ncol];
#pragma unroll
    for (int r = 0; r < 8; ++r) {
      const int m = mbase + r;
      const float v = acc[t][r] + bn;
      out[(size_t)m * 128 + t * 16 + ncol] = 1.0f / (1.0f + __expf(-v));
    }
  }
}

// ---------------- host launch ----------------
extern "C" void kernel_launch(void* const* d_in, const int* in_sizes, int n_in,
                              void* d_out, int out_size, void* d_ws, size_t ws_size,
                              hipStream_t stream) {
  const float*     x  = (const float*)d_in[0];
  const long long* ei = (const long long*)d_in[1];
  const float* W1 = (const float*)d_in[2];
  const float* b1 = (const float*)d_in[3];
  const float* W2 = (const float*)d_in[4];
  const float* b2 = (const float*)d_in[5];
  const float* W3 = (const float*)d_in[6];
  const float* b3 = (const float*)d_in[7];
  const float* Wl = (const float*)d_in[8];
  const float* bl = (const float*)d_in[9];

  const int N = in_sizes[0] / 128;                 // 200000
  const long long E = (long long)in_sizes[1] / 2;  // 6400000
  float* out  = (float*)d_out;

  // workspace: [dinv N][hA N*32][hB N*32]  ~= 52 MB
  float* dinv = (float*)d_ws;
  float* hA   = dinv + N;
  float* hB   = hA + (size_t)N * HH;

  const int T = 256;
  const int nb  = (N + T - 1) / T;
  const unsigned eb = (unsigned)((E + T - 1) / T);
  const unsigned sb = (unsigned)((E * 32 + T - 1) / T);
  const long long n4 = (long long)N * HH / 4;
  const unsigned hb = (unsigned)((n4 + T - 1) / T);
  const int mtiles  = (N + 15) / 16;
  const int gb      = (mtiles + 3) / 4;  // 4 waves (128 threads) per block

  // degrees -> dinv
  deg_init_k<<<nb, T, 0, stream>>>(dinv, N);
  deg_count_k<<<eb, T, 0, stream>>>(ei, dinv, E);
  deg_fin_k<<<nb, T, 0, stream>>>(dinv, N);

  // Layer 1: x[*,128] @ W1 -> hA ; hB = b1 + dinv^2*hA ; scatter ; sigmoid
  gcn_gemm_wmma<128><<<gb, 128, 0, stream>>>(x, W1, b1, dinv, hA, hB, N);
  edge_scatter_k<<<sb, T, 0, stream>>>(ei, dinv, hA, hB, E);
  sigmoid_k<<<hb, T, 0, stream>>>((float4*)hB, n4);

  // Layer 2
  gcn_gemm_wmma<32><<<gb, 128, 0, stream>>>(hB, W2, b2, dinv, hA, hB, N);
  edge_scatter_k<<<sb, T, 0, stream>>>(ei, dinv, hA, hB, E);
  sigmoid_k<<<hb, T, 0, stream>>>((float4*)hB, n4);

  // Layer 3 (no sigmoid)
  gcn_gemm_wmma<32><<<gb, 128, 0, stream>>>(hB, W3, b3, dinv, hA, hB, N);
  edge_scatter_k<<<sb, T, 0, stream>>>(ei, dinv, hA, hB, E);

  // Head: sigmoid(hB @ Wlin + blin) -> d_out
  out_gemm_wmma<<<gb, 128, 0, stream>>>(hB, Wl, bl, out, N);
}